// SpatioTemporalInteractionv2_51762945852130
// MI455X (gfx1250) — compile-verified
//
#include <hip/hip_runtime.h>
#include <hip/hip_bf16.h>

// Problem constants (from reference): T=16, B=128, N=2048, R=T*B=2048.
#define T_STEPS 16
#define B_DIM   128
#define N_DIM   2048
#define R_DIM   2048          // T*B rows of the GEMM
#define BN_CNT  262144        // B*N
#define STR     36            // LDS row stride (32 K + 4 pad): 16B aligned, bank-conflict free

typedef __attribute__((ext_vector_type(2))) float v2f;
typedef __attribute__((ext_vector_type(8))) float v8f;

// ---------------- spatial_sum[t*B+b] = sum_n x[t,b,n] ----------------
__global__ __launch_bounds__(256) void row_sum_kernel(const float* __restrict__ x,
                                                      float* __restrict__ ssum) {
    __shared__ float sdata[256];
    const int p = blockIdx.x;                   // 0..2047  (t*B+b)
    const float* row = x + (size_t)p * N_DIM;
    float s = 0.0f;
    for (int j = threadIdx.x; j < N_DIM; j += 256) s += row[j];
    sdata[threadIdx.x] = s;
    __syncthreads();
    for (int off = 128; off > 0; off >>= 1) {
        if (threadIdx.x < off) sdata[threadIdx.x] += sdata[threadIdx.x + off];
        __syncthreads();
    }
    if (threadIdx.x == 0) ssum[p] = sdata[0];
}

// ---------------- temporal_sum[b*N+n] = sum_t x[t,b,n] ----------------
__global__ __launch_bounds__(256) void col_time_sum_kernel(const float* __restrict__ x,
                                                           float* __restrict__ tsum) {
    const int idx = blockIdx.x * 256 + threadIdx.x;      // 0..BN_CNT-1
    float s = 0.0f;
    #pragma unroll
    for (int t = 0; t < T_STEPS; ++t) s += x[t * BN_CNT + idx];
    tsum[idx] = s;
}

// ---------------- H[r,m] = sum_k X[r,k] * W[m,k]  (fp32 WMMA, double-buffered) ----------------
// Grid (16,16), 256 threads (8 waves). WG tile 128(rows) x 128(cols m).
// Wave (wr=wave/4, wc=wave%4) owns rows [wr*64, +64) x cols [wc*32, +32):
// 4x2 grid of 16x16 accumulators fed by V_WMMA_F32_16X16X4_F32.
// Double buffering: prefetch tile k+1 into VGPRs while WMMAs consume tile k
// from LDS, then store into the other buffer; one barrier per K-iteration.
__global__ __launch_bounds__(256) void gemm_wmma_kernel(const float* __restrict__ X,
                                                        const float* __restrict__ W,
                                                        float* __restrict__ H) {
    __shared__ float Xs[2][128 * STR];
    __shared__ float Ws[2][128 * STR];

    const int tid  = threadIdx.x;
    const int lane = tid & 31;
    const int wave = tid >> 5;
    const int wr   = wave >> 2;          // 0..1
    const int wc   = wave & 3;           // 0..3
    const int row0 = blockIdx.y * 128;
    const int col0 = blockIdx.x * 128;
    const int lrow = lane & 15;          // row/col within 16x16 fragment
    const int lk   = (lane >> 4) << 1;   // K sub-offset: 0 or 2

    // Cooperative load mapping: thread owns col 4*(tid&7) of rows (tid>>3)+32*i.
    const int lr = tid >> 3;             // 0..31
    const int lc = (tid & 7) << 2;       // 0,4,...,28
    const float* Xg = X + (size_t)(row0 + lr) * N_DIM + lc;
    const float* Wg = W + (size_t)(col0 + lr) * N_DIM + lc;

    v8f acc[4][2] = {};                  // zero-init accumulators

    // --- preload tile 0 into LDS buffer 0 ---
    {
        float4 xr[4], wv[4];
        #pragma unroll
        for (int i = 0; i < 4; ++i) {
            xr[i] = *(const float4*)(Xg + (size_t)(i * 32) * N_DIM);
            wv[i] = *(const float4*)(Wg + (size_t)(i * 32) * N_DIM);
        }
        #pragma unroll
        for (int i = 0; i < 4; ++i) {
            *(float4*)(&Xs[0][(lr + 32 * i) * STR + lc]) = xr[i];
            *(float4*)(&Ws[0][(lr + 32 * i) * STR + lc]) = wv[i];
        }
    }
    __syncthreads();

    int cur = 0;
    for (int k0 = 0; k0 < N_DIM; k0 += 32) {
        const int nk = k0 + 32;
        float4 xr[4], wv[4];
        if (nk < N_DIM) {                // prefetch next tile (overlaps WMMAs below)
            #pragma unroll
            for (int i = 0; i < 4; ++i) {
                xr[i] = *(const float4*)(Xg + (size_t)(i * 32) * N_DIM + nk);
                wv[i] = *(const float4*)(Wg + (size_t)(i * 32) * N_DIM + nk);
            }
        }

        const float* Ab = &Xs[cur][(wr * 64 + lrow) * STR + lk];
        const float* Bb = &Ws[cur][(wc * 32 + lrow) * STR + lk];

        #pragma unroll
        for (int kk = 0; kk < 32; kk += 4) {
            const v2f a0 = *(const v2f*)(Ab + 0 * 16 * STR + kk);
            const v2f a1 = *(const v2f*)(Ab + 1 * 16 * STR + kk);
            const v2f a2 = *(const v2f*)(Ab + 2 * 16 * STR + kk);
            const v2f a3 = *(const v2f*)(Ab + 3 * 16 * STR + kk);
            const v2f b0 = *(const v2f*)(Bb + 0 * 16 * STR + kk);
            const v2f b1 = *(const v2f*)(Bb + 1 * 16 * STR + kk);
            acc[0][0] = __builtin_amdgcn_wmma_f32_16x16x4_f32(false, a0, false, b0, (short)0, acc[0][0], false, false);
            acc[0][1] = __builtin_amdgcn_wmma_f32_16x16x4_f32(false, a0, false, b1, (short)0, acc[0][1], false, false);
            acc[1][0] = __builtin_amdgcn_wmma_f32_16x16x4_f32(false, a1, false, b0, (short)0, acc[1][0], false, false);
            acc[1][1] = __builtin_amdgcn_wmma_f32_16x16x4_f32(false, a1, false, b1, (short)0, acc[1][1], false, false);
            acc[2][0] = __builtin_amdgcn_wmma_f32_16x16x4_f32(false, a2, false, b0, (short)0, acc[2][0], false, false);
            acc[2][1] = __builtin_amdgcn_wmma_f32_16x16x4_f32(false, a2, false, b1, (short)0, acc[2][1], false, false);
            acc[3][0] = __builtin_amdgcn_wmma_f32_16x16x4_f32(false, a3, false, b0, (short)0, acc[3][0], false, false);
            acc[3][1] = __builtin_amdgcn_wmma_f32_16x16x4_f32(false, a3, false, b1, (short)0, acc[3][1], false, false);
        }

        if (nk < N_DIM) {                // stage next tile into the other buffer
            const int nxt = cur ^ 1;
            #pragma unroll
            for (int i = 0; i < 4; ++i) {
                *(float4*)(&Xs[nxt][(lr + 32 * i) * STR + lc]) = xr[i];
                *(float4*)(&Ws[nxt][(lr + 32 * i) * STR + lc]) = wv[i];
            }
        }
        __syncthreads();
        cur ^= 1;
    }

    // D layout: VGPR i -> row (i + 8*(lane>>4)), col (lane&15)
    const int rbase = row0 + wr * 64 + ((lane >> 4) << 3);
    const int cbase = col0 + wc * 32 + (lane & 15);
    #pragma unroll
    for (int rb = 0; rb < 4; ++rb)
        #pragma unroll
        for (int cb = 0; cb < 2; ++cb)
            #pragma unroll
            for (int i = 0; i < 8; ++i)
                H[(size_t)(rbase + rb * 16 + i) * N_DIM + cbase + cb * 16] = acc[rb][cb][i];
}

// ---------------- per-feature partial sums over 128-row chunks ----------------
__global__ __launch_bounds__(256) void col_stats_partial_kernel(const float* __restrict__ H,
                                                                float* __restrict__ partS,
                                                                float* __restrict__ partSS) {
    const int m  = blockIdx.x * 256 + threadIdx.x;  // 0..2047
    const int r0 = blockIdx.y * 128;
    float s = 0.0f, ss = 0.0f;
    for (int r = r0; r < r0 + 128; ++r) {
        const float v = H[(size_t)r * N_DIM + m];
        s  += v;
        ss += v * v;
    }
    partS [blockIdx.y * N_DIM + m] = s;
    partSS[blockIdx.y * N_DIM + m] = ss;
}

// ---------------- fold BN stats + gamma/beta into scale/shift ----------------
__global__ __launch_bounds__(256) void bn_finalize_kernel(const float* __restrict__ partS,
                                                          const float* __restrict__ partSS,
                                                          const float* __restrict__ gamma,
                                                          const float* __restrict__ beta,
                                                          float* __restrict__ scaleA,
                                                          float* __restrict__ shiftA) {
    const int m = blockIdx.x * 256 + threadIdx.x;
    float s = 0.0f, ss = 0.0f;
    #pragma unroll
    for (int j = 0; j < 16; ++j) {
        s  += partS [j * N_DIM + m];
        ss += partSS[j * N_DIM + m];
    }
    const float mu  = s  * (1.0f / (float)R_DIM);
    const float var = ss * (1.0f / (float)R_DIM) - mu * mu;
    const float sc  = gamma[m] * rsqrtf(var + 1e-5f);
    scaleA[m] = sc;
    shiftA[m] = beta[m] - mu * sc;
}

// ---------------- fused BN-apply + 3 LIF chains + residual output ----------------
__global__ __launch_bounds__(256) void lif_fuse_kernel(const float* __restrict__ x,
                                                       const float* __restrict__ H,
                                                       const float* __restrict__ ssum,
                                                       const float* __restrict__ tsum,
                                                       const float* __restrict__ scaleA,
                                                       const float* __restrict__ shiftA,
                                                       float* __restrict__ out) {
    const int idx = blockIdx.x * 256 + threadIdx.x;  // b*N + m, 0..BN_CNT-1
    const int b = idx >> 11;
    const int m = idx & (N_DIM - 1);
    const float sc = scaleA[m];
    const float sh = shiftA[m];
    const float ts = tsum[idx];

    float v1 = 0.0f, v2 = 0.0f, v3 = 0.0f;
    #pragma unroll
    for (int t = 0; t < T_STEPS; ++t) {
        const int off = t * BN_CNT + idx;
        const float h = H[off] * sc + sh;           // BatchNorm
        // LIF on h:  v = v/2 + h ; fire ; hard reset
        v1 = v1 * 0.5f + h;
        const float s1 = (v1 >= 1.0f) ? 1.0f : 0.0f;
        v1 = (v1 >= 1.0f) ? 0.0f : v1;
        // spatial LIF on s1 * spatial_sum[t,b]
        const float ssv = ssum[t * B_DIM + b];
        v2 = v2 * 0.5f + s1 * ssv;
        const float s2 = (v2 >= 1.0f) ? 1.0f : 0.0f;
        v2 = (v2 >= 1.0f) ? 0.0f : v2;
        // temporal LIF on s1 * temporal_sum[b,m]
        v3 = v3 * 0.5f + s1 * ts;
        const float s3 = (v3 >= 1.0f) ? 1.0f : 0.0f;
        v3 = (v3 >= 1.0f) ? 0.0f : v3;

        out[off] = x[off] + s2 * s3;
    }
}

extern "C" void kernel_launch(void* const* d_in, const int* in_sizes, int n_in,
                              void* d_out, int out_size, void* d_ws, size_t ws_size,
                              hipStream_t stream) {
    const float* x     = (const float*)d_in[0];   // [16,128,2048]
    const float* W     = (const float*)d_in[1];   // [2048,2048]
    const float* gamma = (const float*)d_in[2];   // [2048]
    const float* beta  = (const float*)d_in[3];   // [2048]
    float* out = (float*)d_out;

    // Workspace layout (floats)
    float* ws     = (float*)d_ws;
    float* H      = ws;                           // 2048*2048
    float* ssum   = H      + (size_t)R_DIM * N_DIM; // 2048
    float* tsum   = ssum   + R_DIM;               // 262144
    float* partS  = tsum   + BN_CNT;              // 16*2048
    float* partSS = partS  + 16 * N_DIM;          // 16*2048
    float* scaleA = partSS + 16 * N_DIM;          // 2048
    float* shiftA = scaleA + N_DIM;               // 2048

    row_sum_kernel<<<R_DIM, 256, 0, stream>>>(x, ssum);
    col_time_sum_kernel<<<BN_CNT / 256, 256, 0, stream>>>(x, tsum);
    gemm_wmma_kernel<<<dim3(N_DIM / 128, R_DIM / 128), 256, 0, stream>>>(x, W, H);
    col_stats_partial_kernel<<<dim3(N_DIM / 256, 16), 256, 0, stream>>>(H, partS, partSS);
    bn_finalize_kernel<<<N_DIM / 256, 256, 0, stream>>>(partS, partSS, gamma, beta, scaleA, shiftA);
    lif_fuse_kernel<<<BN_CNT / 256, 256, 0, stream>>>(x, H, ssum, tsum, scaleA, shiftA, out);
}